// GeometricHashingClass_20418274525829
// MI455X (gfx1250) — compile-verified
//
#include <hip/hip_runtime.h>
#include <math.h>

// Geometric hashing: pure streaming elementwise map.
//   in : keypointCoordinates [B,S,3,2] f32 (24 B/item), pixelCoordinates [B,S,2] f32 (8 B/item)
//   out: [B,S,2] f32 (8 B/item)
// Memory-bound: ~80 MB total traffic -> ~3.4 us floor at 23.3 TB/s.
// Strategy: 2 items / "pair" so every access is an aligned 128-bit vector op,
// 2 pairs per thread for memory-level parallelism, non-temporal hints since
// data is touched exactly once.

typedef __attribute__((ext_vector_type(4))) float v4f;

struct f2 { float x, y; };

__device__ __forceinline__ f2 geo_hash_one(
    float x0, float y0, float x1, float y1, float x2, float y2,
    float pix_x, float pix_y)
{
    // keypointSwap(kp, 2, 1, Y): cond = kp2.y > kp1.y ; kp2 = cond ? kp[1] : kp[0]
    {
        bool c = y2 > y1;
        float nx = c ? x1 : x0;
        float ny = c ? y1 : y0;
        x2 = nx; y2 = ny;
    }
    // keypointSwap(kp, 2, 0, Y): cond = kp2.y > kp0.y ; kp2 = cond ? kp[1] : kp[0]
    {
        bool c = y2 > y0;
        float nx = c ? x1 : x0;
        float ny = c ? y1 : y0;
        x2 = nx; y2 = ny;
    }
    // keypointSwap(kp, 0, 1, X): cond = kp0.x > kp1.x ; kp0 = cond ? kp[1] : kp[0]
    {
        bool c = x0 > x1;
        float nx = c ? x1 : x0;
        float ny = c ? y1 : y0;
        x0 = nx; y0 = ny;
    }

    // step1: translate so triangle base centre is at origin
    float bcx = (x0 + x1) * 0.5f;
    float bcy = (y0 + y1) * 0.5f;
    float pxv = pix_x - bcx;
    float pyv = pix_y - bcy;

    // step2: rotate base vector onto x axis
    float bvx = x1 - x0;
    float bvy = y1 - y0;
    float nrm = sqrtf(bvx * bvx + bvy * bvy);
    float c = bvx / nrm;
    float s = bvy / nrm;
    float px = c * pxv + s * pyv;
    float py = -s * pxv + c * pyv;

    // step3: scale x by base size
    float base_size_x = x0 - x1;
    px = px / base_size_x;

    // step4: scale y by triangle height
    float height = y2 - y1;
    py = py / height;

    // step5: shear to remove tip x-offset (base_centre_x == bcx)
    float tip_vec_x = x2 - bcx;
    float tip_vec_y = y2 - y0;
    float shear = tip_vec_x / tip_vec_y;
    px = px - shear * py;

    f2 r;
    r.x = px;
    r.y = py;
    return r;
}

#define PAIRS_PER_THREAD 2

__global__ __launch_bounds__(256) void geohash_pairs_kernel(
    const v4f* __restrict__ kp4,   // 3 x v4f per pair (48 B = 2 items of kp)
    const v4f* __restrict__ pix4,  // 1 x v4f per pair (2 items of pix)
    v4f* __restrict__ out4,        // 1 x v4f per pair (2 items of output)
    int npairs)
{
    int base = blockIdx.x * (blockDim.x * PAIRS_PER_THREAD) + threadIdx.x;
#pragma unroll
    for (int it = 0; it < PAIRS_PER_THREAD; ++it) {
        int p = base + it * blockDim.x;
        if (p < npairs) {
            // item A keypoints: a.xyzw = (x0,y0,x1,y1), b.xy = (x2,y2)
            // item B keypoints: b.zw = (x0,y0), k.xyzw = (x1,y1,x2,y2)
            v4f a = __builtin_nontemporal_load(&kp4[3 * p + 0]);
            v4f b = __builtin_nontemporal_load(&kp4[3 * p + 1]);
            v4f k = __builtin_nontemporal_load(&kp4[3 * p + 2]);
            v4f q = __builtin_nontemporal_load(&pix4[p]);

            f2 rA = geo_hash_one(a.x, a.y, a.z, a.w, b.x, b.y, q.x, q.y);
            f2 rB = geo_hash_one(b.z, b.w, k.x, k.y, k.z, k.w, q.z, q.w);

            v4f r;
            r.x = rA.x; r.y = rA.y; r.z = rB.x; r.w = rB.y;
            __builtin_nontemporal_store(r, &out4[p]);
        }
    }
}

// Scalar tail for an odd final item (not hit for B=128,S=16384, kept for generality).
__global__ void geohash_tail_kernel(
    const float* __restrict__ kp,
    const float* __restrict__ pix,
    float* __restrict__ out,
    int start, int n_items)
{
    int i = start + blockIdx.x * blockDim.x + threadIdx.x;
    if (i < n_items) {
        const float* kk = kp + 6 * i;
        f2 r = geo_hash_one(kk[0], kk[1], kk[2], kk[3], kk[4], kk[5],
                            pix[2 * i], pix[2 * i + 1]);
        out[2 * i]     = r.x;
        out[2 * i + 1] = r.y;
    }
}

extern "C" void kernel_launch(void* const* d_in, const int* in_sizes, int n_in,
                              void* d_out, int out_size, void* d_ws, size_t ws_size,
                              hipStream_t stream) {
    (void)n_in; (void)out_size; (void)d_ws; (void)ws_size;

    const float* kp  = (const float*)d_in[0];   // [B,S,3,2]
    const float* pix = (const float*)d_in[1];   // [B,S,2]
    float* out = (float*)d_out;                 // [B,S,2]

    const int n_items = in_sizes[1] / 2;        // B*S
    const int npairs  = n_items / 2;

    const int threads = 256;
    const int pairs_per_block = threads * PAIRS_PER_THREAD;
    const int blocks = (npairs + pairs_per_block - 1) / pairs_per_block;

    if (blocks > 0) {
        geohash_pairs_kernel<<<blocks, threads, 0, stream>>>(
            (const v4f*)kp, (const v4f*)pix, (v4f*)out, npairs);
    }
    if (n_items & 1) {
        geohash_tail_kernel<<<1, 64, 0, stream>>>(kp, pix, out, n_items - 1, n_items);
    }
}